// Integrator_82463372083721
// MI455X (gfx1250) — compile-verified
//
#include <hip/hip_runtime.h>

// ---------------------------------------------------------------------------
// Volumetric scatter-accumulate integrator for MI455X (gfx1250).
//
//  * Accumulators (vcache -> d_out[0:N), wcache -> d_out[N:2N), 2MB touched
//    bitmask in d_ws) ~130MB total: fits the 192MB L2; all 50M atomics
//    resolve on-chip as no-return global_atomic_add_f32 / atomic_or.
//  * Index stream (192MB, AoS, 96B/point) staged per 256-point tile into LDS
//    via coalesced GLOBAL_LOAD_ASYNC_TO_LDS_B128 (ASYNCcnt), double-buffered
//    so the next tile's DMA overlaps the current tile's atomics.
//  * Weights/values streamed with non-temporal loads; outputs stored NT.
// ---------------------------------------------------------------------------

typedef float __attribute__((ext_vector_type(4))) f32x4;
typedef int   __attribute__((ext_vector_type(4))) i32x4;
typedef unsigned int u32;

#define VOLD 256
#define NVOX (VOLD * VOLD * VOLD)      // 16,777,216 voxels
#define NPTS (256 * 256 * 32)          // 2,097,152 sample points
#define THREADS 256
#define TILE 256                       // points per LDS tile
#define NTILES (NPTS / TILE)           // 8192
#define IDX_TILE_BYTES (TILE * 96)     // 24576 B of indices per tile
#define SCATTER_BLOCKS 2048

#ifndef __has_builtin
#define __has_builtin(x) 0
#endif

#if __has_builtin(__builtin_amdgcn_global_load_async_to_lds_b128)
#define ASYNC_BUILTIN 1
#else
#define ASYNC_BUILTIN 0
#endif

// Builtin signature (from clang diagnostic): param0 = v4i in addrspace(1)
// ("__device__"), param1 = v4i in addrspace(3) ("__shared__"), then imm
// offset and imm cpol.
typedef __attribute__((address_space(1))) i32x4 as1_i32x4;
typedef __attribute__((address_space(3))) i32x4 as3_i32x4;

__device__ __forceinline__ void async_copy_b128(const void* g, void* l) {
#if ASYNC_BUILTIN
    __builtin_amdgcn_global_load_async_to_lds_b128(
        (as1_i32x4*)g, (as3_i32x4*)l, 0, 0);
#else
    const u32 loff = (u32)(unsigned long long)l;   // LDS offset = low 32 bits
    asm volatile("global_load_async_to_lds_b128 %0, %1, off"
                 :: "v"(loff), "v"(g) : "memory");
#endif
}

__device__ __forceinline__ void wait_async_0() {
#if __has_builtin(__builtin_amdgcn_s_wait_asynccnt)
    __builtin_amdgcn_s_wait_asynccnt(0);
#else
    asm volatile("s_wait_asynccnt 0x0" ::: "memory");
#endif
}

__device__ __forceinline__ void wait_async_6() {  // one 6-op tile may remain
#if __has_builtin(__builtin_amdgcn_s_wait_asynccnt)
    __builtin_amdgcn_s_wait_asynccnt(6);
#else
    asm volatile("s_wait_asynccnt 0x6" ::: "memory");
#endif
}

__device__ __forceinline__ void atomic_add_f32_nr(float* p, float v) {
    (void)__hip_atomic_fetch_add(p, v, __ATOMIC_RELAXED, __HIP_MEMORY_SCOPE_AGENT);
}

// --------------------------- scatter (main) --------------------------------
__global__ void __launch_bounds__(THREADS)
integ_scatter(const float* __restrict__ values,
              const int*   __restrict__ indices,
              const float* __restrict__ weights,
              float* __restrict__ vcache,   // d_out[0:N)
              float* __restrict__ wcache,   // d_out[N:2N)
              u32*   __restrict__ mask) {
    __shared__ alignas(16) char sIdx[2][IDX_TILE_BYTES];   // 48KB double buffer

    const int t = threadIdx.x;

    // Issue coalesced async DMA of one tile's 24KB index block into LDS buf.
    auto issue_tile = [&](int tile, int buf) {
        const char* g = (const char*)indices + (size_t)tile * IDX_TILE_BYTES;
#pragma unroll
        for (int k = 0; k < 6; ++k) {
            const int off = k * 4096 + t * 16;
            async_copy_b128(g + off, sIdx[buf] + off);
        }
    };

    int tile = blockIdx.x;
    int buf  = 0;
    issue_tile(tile, buf);

    for (; tile < NTILES; tile += gridDim.x) {
        const int next = tile + gridDim.x;
        if (next < NTILES) {
            issue_tile(next, buf ^ 1);   // overlap next tile's DMA
            wait_async_6();              // current tile's 6 ops complete
        } else {
            wait_async_0();
        }
        __syncthreads();                 // tile data visible to all waves

        const int p = tile * TILE + t;

        const float v = __builtin_nontemporal_load(values + p);
        const f32x4* wp = (const f32x4*)(weights + (size_t)p * 8);
        const f32x4 w0 = __builtin_nontemporal_load(wp + 0);
        const f32x4 w1 = __builtin_nontemporal_load(wp + 1);
        // prefetch next tile's weights (global_prefetch_b8)
        if (next < NTILES) {
            __builtin_prefetch(weights + (size_t)(next * TILE + t) * 8, 0, 1);
        }

        const i32x4* lp = (const i32x4*)(sIdx[buf] + t * 96);
        int a[24];
#pragma unroll
        for (int k = 0; k < 6; ++k) {
            const i32x4 iv = lp[k];
            a[4 * k + 0] = iv[0];
            a[4 * k + 1] = iv[1];
            a[4 * k + 2] = iv[2];
            a[4 * k + 3] = iv[3];
        }
        const float ws[8] = {w0[0], w0[1], w0[2], w0[3],
                             w1[0], w1[1], w1[2], w1[3]};

#pragma unroll
        for (int c = 0; c < 8; ++c) {
            const int x = a[3 * c + 0];
            const int y = a[3 * c + 1];
            const int z = a[3 * c + 2];
            const bool valid = ((u32)x < (u32)VOLD) &&
                               ((u32)y < (u32)VOLD) &&
                               ((u32)z < (u32)VOLD);
            if (valid) {
                const int flat = x * (VOLD * VOLD) + y * VOLD + z;
                const float w = ws[c];
                atomic_add_f32_nr(vcache + flat, w * v);
                atomic_add_f32_nr(wcache + flat, w);
                atomicOr(mask + (flat >> 5), 1u << (flat & 31));
            }
        }

        __syncthreads();                 // all reads done before buf is reused
        buf ^= 1;
    }
}

// ------------------------------- init --------------------------------------
__global__ void integ_init(float* __restrict__ out, u32* __restrict__ mask) {
    const int t = blockIdx.x * blockDim.x + threadIdx.x;    // [0, NVOX/4)
    const f32x4 z = {0.f, 0.f, 0.f, 0.f};
    ((f32x4*)out)[t]            = z;
    ((f32x4*)out)[NVOX / 4 + t] = z;
    if (t < NVOX / 128) {                                   // 2MB bitmask
        ((f32x4*)mask)[t] = z;
    }
}

// ----------------------------- finalize ------------------------------------
__global__ void integ_finalize(const float* __restrict__ values_volume,
                               const float* __restrict__ weights_volume,
                               float* __restrict__ out,     // in-place
                               const u32* __restrict__ mask) {
    const int t = blockIdx.x * blockDim.x + threadIdx.x;    // [0, NVOX/4)
    const f32x4 vc = ((const f32x4*)out)[t];
    const f32x4 wc = ((const f32x4*)out)[NVOX / 4 + t];
    const f32x4 vv = __builtin_nontemporal_load((const f32x4*)values_volume + t);
    const f32x4 wv = __builtin_nontemporal_load((const f32x4*)weights_volume + t);

    const u32 m    = mask[t >> 3];
    const u32 bits = (m >> ((t & 7) * 4)) & 0xFu;

    f32x4 newv, neww;
#pragma unroll
    for (int k = 0; k < 4; ++k) {
        const float denom   = wv[k] + wc[k];
        const bool  touched = (bits >> k) & 1u;
        newv[k] = touched ? (wv[k] * vv[k] + vc[k]) / denom : vv[k];
        neww[k] = touched ? denom : wv[k];
    }
    __builtin_nontemporal_store(newv, (f32x4*)out + t);
    __builtin_nontemporal_store(neww, (f32x4*)out + NVOX / 4 + t);
}

// ------------------------------ launcher -----------------------------------
extern "C" void kernel_launch(void* const* d_in, const int* in_sizes, int n_in,
                              void* d_out, int out_size, void* d_ws, size_t ws_size,
                              hipStream_t stream) {
    const float* values         = (const float*)d_in[0];  // [1, 65536, 32]
    const int*   indices        = (const int*)  d_in[1];  // [1, 65536, 32, 8, 3]
    const float* weights        = (const float*)d_in[2];  // [1, 65536, 32, 8]
    const float* values_volume  = (const float*)d_in[3];  // [256,256,256]
    const float* weights_volume = (const float*)d_in[4];  // [256,256,256]

    float* out  = (float*)d_out;       // [0:N) new_values, [N:2N) new_weights
    u32*   mask = (u32*)d_ws;          // 2MB touched bitmask

    const int vox_blocks = (NVOX / 4) / THREADS;           // 16384

    integ_init<<<vox_blocks, THREADS, 0, stream>>>(out, mask);

    integ_scatter<<<SCATTER_BLOCKS, THREADS, 0, stream>>>(
        values, indices, weights, out, out + NVOX, mask);

    integ_finalize<<<vox_blocks, THREADS, 0, stream>>>(
        values_volume, weights_volume, out, mask);
}